// BayesianLayer_63170378989673
// MI455X (gfx1250) — compile-verified
//
#include <hip/hip_runtime.h>
#include <math.h>

#define B_    128
#define IN_   1024
#define OUT_  1024
#define ISPLIT 4
#define ICH   (IN_ / ISPLIT)   // 256 i per wave
#define OCH   128              // o per wave (32 lanes x float4)

typedef __attribute__((ext_vector_type(2))) float v2f;
typedef __attribute__((ext_vector_type(4))) float v4f;
typedef __attribute__((ext_vector_type(8))) float v8f;

// ---------------------------------------------------------------------------
// Kernel 1: sigma = softplus(ro) precompute (4 MB, written once, L2-resident)
// ---------------------------------------------------------------------------
__global__ void __launch_bounds__(256) softplus_kernel(const float* __restrict__ ro,
                                                       float* __restrict__ sigma) {
    int idx = blockIdx.x * 256 + threadIdx.x;
    sigma[idx] = log1pf(expf(ro[idx]));
}

// ---------------------------------------------------------------------------
// Kernel 2: out[b,o] = (x @ mu)[b,o] + eps_bias[b,o]*softplus(ro_bias[o]) + mu_bias[o]
// fp32 WMMA 16x16x4: one wave per 16x16 output tile (8 b-tiles x 64 o-tiles).
// A 16x4 layout:  lane = m + 16*(k>=2), vgpr = k&1
// B 4x16 layout:  lane = n + 16*(k>=2), vgpr = k&1
// C 16x16 layout: lane = n + 16*(m>=8), vgpr = m&7
// ---------------------------------------------------------------------------
__global__ void __launch_bounds__(256) wmma_mu_bias_kernel(const float* __restrict__ x,
                                                           const float* __restrict__ mu,
                                                           const float* __restrict__ eps_bias,
                                                           const float* __restrict__ mu_bias,
                                                           const float* __restrict__ ro_bias,
                                                           float* __restrict__ out) {
    const int lane = threadIdx.x & 31;
    const int wid  = (blockIdx.x * blockDim.x + threadIdx.x) >> 5;  // 0..511
    const int to   = wid & 63;                                      // o-tile
    const int tb   = wid >> 6;                                      // b-tile
    const int o0   = to * 16;
    const int b0   = tb * 16;
    const int n    = lane & 15;          // N index (and M index for A rows)
    const int kh   = (lane >> 4) << 1;   // 0 for lanes 0-15, 2 for lanes 16-31

    const float* __restrict__ xrow = x + (size_t)(b0 + n) * IN_;

    v8f c0 = {};
    v8f c1 = {};
    for (int k0 = 0; k0 < IN_; k0 += 8) {
        v2f a0, b0v, a1, b1v;
        a0.x  = xrow[k0 + kh];
        a0.y  = xrow[k0 + kh + 1];
        b0v.x = mu[(size_t)(k0 + kh) * OUT_ + o0 + n];
        b0v.y = mu[(size_t)(k0 + kh + 1) * OUT_ + o0 + n];
        a1.x  = xrow[k0 + 4 + kh];
        a1.y  = xrow[k0 + 4 + kh + 1];
        b1v.x = mu[(size_t)(k0 + 4 + kh) * OUT_ + o0 + n];
        b1v.y = mu[(size_t)(k0 + 4 + kh + 1) * OUT_ + o0 + n];
        // two accumulators to break the WMMA->WMMA RAW dependency chain
        c0 = __builtin_amdgcn_wmma_f32_16x16x4_f32(false, a0, false, b0v,
                                                   (short)0, c0, false, false);
        c1 = __builtin_amdgcn_wmma_f32_16x16x4_f32(false, a1, false, b1v,
                                                   (short)0, c1, false, false);
    }

    const float sb  = log1pf(expf(ro_bias[o0 + n]));
    const float mb  = mu_bias[o0 + n];
    const int   mhi = (lane >> 4) << 3;  // +8 rows for upper lane half
#pragma unroll
    for (int r = 0; r < 8; ++r) {
        const int bb = b0 + mhi + r;
        const int oo = o0 + n;
        out[(size_t)bb * OUT_ + oo] = c0[r] + c1[r] + eps_bias[(size_t)bb * OUT_ + oo] * sb + mb;
    }
}

// ---------------------------------------------------------------------------
// Kernel 3: bandwidth-bound stream of eps (512 MB).
// One wave per (b, o-chunk of 128, i-chunk of 256): 4096 waves, each issuing
// unrolled b128 loads (~16 MB in flight device-wide) to cover HBM latency.
// eps is single-use -> non-temporal loads so the 512 MB stream does not evict
// the hot, 128x-reused sigma/mu/x working set from the 192 MB L2.
// partial[ic][b][o] = sum_{i in chunk} x[b,i] * eps[b,i,o] * sigma[i,o]
// ---------------------------------------------------------------------------
__global__ void __launch_bounds__(256) eps_stream_kernel(const float* __restrict__ x,
                                                         const float* __restrict__ eps,
                                                         const float* __restrict__ sigma,
                                                         float* __restrict__ partial) {
    const int lane = threadIdx.x & 31;
    const int wid  = (blockIdx.x * blockDim.x + threadIdx.x) >> 5;  // 0..4095
    const int ic   = wid & (ISPLIT - 1);
    const int oc   = (wid >> 2) & 7;
    const int b    = wid >> 5;
    const int o    = oc * OCH + lane * 4;
    const int i0   = ic * ICH;

    const v4f* __restrict__ ep =
        reinterpret_cast<const v4f*>(eps) + ((((size_t)b * IN_ + i0) * OUT_ + o) >> 2);
    const v4f* __restrict__ sg =
        reinterpret_cast<const v4f*>(sigma) + ((((size_t)i0) * OUT_ + o) >> 2);
    const float* __restrict__ xb = x + (size_t)b * IN_ + i0;

    float ax = 0.f, ay = 0.f, az = 0.f, aw = 0.f;
#pragma unroll 8
    for (int i = 0; i < ICH; ++i) {
        const v4f   e  = __builtin_nontemporal_load(ep + (size_t)i * (OUT_ / 4));
        const v4f   s  = sg[(size_t)i * (OUT_ / 4)];
        const float xv = xb[i];
        ax += xv * (e.x * s.x);
        ay += xv * (e.y * s.y);
        az += xv * (e.z * s.z);
        aw += xv * (e.w * s.w);
    }
    v4f r;
    r.x = ax; r.y = ay; r.z = az; r.w = aw;
    reinterpret_cast<v4f*>(partial)[((((size_t)ic * B_ + b) * OUT_) + o) >> 2] = r;
}

// ---------------------------------------------------------------------------
// Kernel 4: out += sum over the 4 i-chunk partials
// ---------------------------------------------------------------------------
__global__ void __launch_bounds__(256) reduce_kernel(const float* __restrict__ partial,
                                                     float* __restrict__ out) {
    const size_t idx = (size_t)blockIdx.x * 256 + threadIdx.x;  // over B_*OUT_
    const size_t N   = (size_t)B_ * OUT_;
    float s = partial[idx] + partial[idx + N] + partial[idx + 2 * N] + partial[idx + 3 * N];
    out[idx] += s;
}

// ---------------------------------------------------------------------------
// Fallback (workspace too small): single-pass stream, softplus inline.
// 1024 waves, one per (b, o-chunk of 128). out += accumulation.
// ---------------------------------------------------------------------------
__global__ void __launch_bounds__(256) eps_fallback_kernel(const float* __restrict__ x,
                                                           const float* __restrict__ eps,
                                                           const float* __restrict__ ro,
                                                           float* __restrict__ out) {
    const int lane = threadIdx.x & 31;
    const int wid  = (blockIdx.x * blockDim.x + threadIdx.x) >> 5;  // 0..1023
    const int oc   = wid & 7;
    const int b    = wid >> 3;
    const int o    = oc * OCH + lane * 4;

    const v4f* __restrict__ ep =
        reinterpret_cast<const v4f*>(eps) + ((((size_t)b * IN_) * OUT_ + o) >> 2);
    const v4f* __restrict__ rp =
        reinterpret_cast<const v4f*>(ro) + ((size_t)o >> 2);
    const float* __restrict__ xb = x + (size_t)b * IN_;

    float ax = 0.f, ay = 0.f, az = 0.f, aw = 0.f;
#pragma unroll 4
    for (int i = 0; i < IN_; ++i) {
        const v4f   e  = __builtin_nontemporal_load(ep + (size_t)i * (OUT_ / 4));
        const v4f   rr = rp[(size_t)i * (OUT_ / 4)];
        const float xv = xb[i];
        ax += xv * (e.x * log1pf(expf(rr.x)));
        ay += xv * (e.y * log1pf(expf(rr.y)));
        az += xv * (e.z * log1pf(expf(rr.z)));
        aw += xv * (e.w * log1pf(expf(rr.w)));
    }
    v4f* op = reinterpret_cast<v4f*>(out) + ((((size_t)b * OUT_) + o) >> 2);
    v4f v = *op;
    v.x += ax; v.y += ay; v.z += az; v.w += aw;
    *op = v;
}

extern "C" void kernel_launch(void* const* d_in, const int* in_sizes, int n_in,
                              void* d_out, int out_size, void* d_ws, size_t ws_size,
                              hipStream_t stream) {
    const float* x        = (const float*)d_in[0];  // [B, IN]
    const float* eps      = (const float*)d_in[1];  // [B, IN, OUT]
    const float* eps_bias = (const float*)d_in[2];  // [B, OUT]
    const float* mu       = (const float*)d_in[3];  // [IN, OUT]
    const float* ro       = (const float*)d_in[4];  // [IN, OUT]
    const float* mu_bias  = (const float*)d_in[5];  // [1, OUT]
    const float* ro_bias  = (const float*)d_in[6];  // [1, OUT]
    float* out = (float*)d_out;

    // Step 1: out = x @ mu + bias  (fp32 WMMA, 512 waves -> 64 blocks x 256)
    wmma_mu_bias_kernel<<<64, 256, 0, stream>>>(x, mu, eps_bias, mu_bias, ro_bias, out);

    const size_t need = ((size_t)IN_ * OUT_ + (size_t)ISPLIT * B_ * OUT_) * sizeof(float);
    if (ws_size >= need) {
        float* sigma   = (float*)d_ws;                     // [IN, OUT]
        float* partial = sigma + (size_t)IN_ * OUT_;       // [ISPLIT, B, OUT]
        // Step 2: sigma = softplus(ro)
        softplus_kernel<<<(IN_ * OUT_) / 256, 256, 0, stream>>>(ro, sigma);
        // Step 3: stream eps (4096 waves -> 512 blocks x 256), eps loads NT
        eps_stream_kernel<<<512, 256, 0, stream>>>(x, eps, sigma, partial);
        // Step 4: out += reduce(partials)
        reduce_kernel<<<(B_ * OUT_) / 256, 256, 0, stream>>>(partial, out);
    } else {
        // Workspace-free fallback: inline softplus, single pass.
        eps_fallback_kernel<<<(1024 * 32) / 256, 256, 0, stream>>>(x, eps, ro, out);
    }
}